// QuantumTTEmbedding_55886114455743
// MI455X (gfx1250) — compile-verified
//
#include <hip/hip_runtime.h>

typedef __attribute__((ext_vector_type(2))) float v2f;
typedef __attribute__((ext_vector_type(4))) float v4f;
typedef __attribute__((ext_vector_type(8))) float v8f;

#define V1c 20
#define V2c 40
#define V3c 40

// LDS layout (float offsets)
#define OFF_CR1 0            // 320   = 20*4*4
#define OFF_CR2 320          // 2560  = 4*40*4*4
#define OFF_CR3 2880         // 1280  = 4*40*8
#define OFF_CI1 4160
#define OFF_CI2 4480
#define OFF_CI3 7040
#define OFF_WAVE 8320        // + wave*512 : [0..127] tmpT (upper half scratch),
                             //              [128..383] out staging, [384..511] dump
#define LDS_FLOATS (8320 + 8 * 512)   // 12416 floats = 49.7 KB

__global__ __launch_bounds__(256) void tt_embed_wmma_kernel(
    const int* __restrict__ ids,
    const float* __restrict__ cr1, const float* __restrict__ cr2, const float* __restrict__ cr3,
    const float* __restrict__ ci1, const float* __restrict__ ci2, const float* __restrict__ ci3,
    float* __restrict__ out, int ntok)
{
    __shared__ __align__(16) float lds[LDS_FLOATS];
    const int tid = threadIdx.x;

    // Cooperative load of all six TT cores into LDS (~33 KB, L2-hot).
    for (int x = tid; x < 320;  x += 256) { lds[OFF_CR1 + x] = cr1[x]; lds[OFF_CI1 + x] = ci1[x]; }
    for (int x = tid; x < 2560; x += 256) { lds[OFF_CR2 + x] = cr2[x]; lds[OFF_CI2 + x] = ci2[x]; }
    for (int x = tid; x < 1280; x += 256) { lds[OFF_CR3 + x] = cr3[x]; lds[OFF_CI3 + x] = ci3[x]; }
    __syncthreads();

    const int lane  = tid & 31;
    const int wv    = __builtin_amdgcn_readfirstlane(tid >> 5);  // wave id (uniform)
    const int khalf = lane >> 4;   // 0: K={0,1}, 1: K={2,3} (f32 WMMA operand layout)
    const int mn    = lane & 15;   // M (A operand) / N (B operand) within 16x16 tile

    float* wbase = lds + OFF_WAVE + wv * 512;
    float* tmpT  = wbase;          // tmpT[n*4 + m] = temp[m][n] (m=0..3 valid)
    float* ostg  = wbase + 128;    // 256 floats, final output layout

    // Branch-free operand masks / clamped indices
    const bool mlt4 = (mn < 4);
    const bool nlt8 = (mn < 8);
    const int  m4   = mn & 3;      // clamped row for A1 load
    const int  e8   = mn & 7;      // clamped col for B2 load
    const int  r0   = 2 * khalf;

    // D2 staging address: valid lanes -> final layout slot; invalid -> dump (collision-free)
    const int stg_valid_base = 128 + khalf * 64 + mn;        // + h*128 added in loop
    const int stg_dump_base  = 384 + khalf * 64 + e8;

    const int nwaves = gridDim.x * 8;
    for (int t = blockIdx.x * 8 + wv; t < ntok; t += nwaves) {
        // --- factor the token id (wave-uniform -> scalar path) ---
        int id  = __builtin_amdgcn_readfirstlane(ids[t]);
        int idx = id % (V1c * V2c * V3c);
        int kk  = idx % V3c;
        int t2  = idx / V3c;
        int jj  = t2 % V2c;
        int ii  = t2 / V2c;

        v8f acc = {};  // zero accumulator

#pragma unroll
        for (int h = 0; h < 2; ++h) {   // h=0: real, h=1: imag
            const float* g1  = lds + (h ? OFF_CI1 : OFF_CR1) + ii * 16; // [d][r], 4x4
            const float* c2p = lds + (h ? OFF_CI2 : OFF_CR2);           // [r][40][f][s]
            const float* c3p = lds + (h ? OFF_CI3 : OFF_CR3);           // [s][40][e]

            // ---- WMMA #1: temp(16x16) = A1(16x4) * B1(4x16), rows 0..3 of M valid
            // Unconditional clamped loads + VALU select (no exec-mask branches).
            float a1x = g1[m4 * 4 + r0    ];
            float a1y = g1[m4 * 4 + r0 + 1];
            v2f a1;
            a1.x = mlt4 ? a1x : 0.0f;
            a1.y = mlt4 ? a1y : 0.0f;
            // B layout: lane holds col N=mn; B1[r][f*4+s] = c2[r][jj][f][s]
            const int f1 = mn >> 2, s1 = mn & 3;
            v2f b1;
            b1.x = c2p[(((r0    ) * V2c + jj) * 4 + f1) * 4 + s1];
            b1.y = c2p[(((r0 + 1) * V2c + jj) * 4 + f1) * 4 + s1];
            v8f d1v = __builtin_amdgcn_wmma_f32_16x16x4_f32(
                false, a1, false, b1, (short)0, acc, false, false);

            // Stage temp transposed; ALL lanes store b128 (lanes 16..31 hit scratch half)
            {
                v4f st; st.x = d1v[0]; st.y = d1v[1]; st.z = d1v[2]; st.w = d1v[3];
                *(v4f*)(tmpT + lane * 4) = st;
            }
            __builtin_amdgcn_wave_barrier();

            // ---- WMMA #2: final(16x16) = A2(16x4) * B2(4x16), cols 0..7 of N valid
            // A2[df][s] = temp[df>>2][(df&3)*4 + s]
            const int d_ = mn >> 2, f_ = mn & 3;
            const int na = f_ * 4 + r0;
            v2f a2;
            a2.x = tmpT[ na      * 4 + d_];
            a2.y = tmpT[(na + 1) * 4 + d_];
            // B2[s][e] = c3[s][kk][e]; unconditional clamped load + select
            float b2x = c3p[((r0    ) * V3c + kk) * 8 + e8];
            float b2y = c3p[((r0 + 1) * V3c + kk) * 8 + e8];
            v2f b2;
            b2.x = nlt8 ? b2x : 0.0f;
            b2.y = nlt8 ? b2y : 0.0f;
            v8f d2v = __builtin_amdgcn_wmma_f32_16x16x4_f32(
                false, a2, false, b2, (short)0, acc, false, false);

            // Stage into final output order: pos = (v + 8*khalf)*8 + e (per half);
            // invalid lanes write to the dump region -> all stores unconditional.
            {
                float* ob = wbase + (nlt8 ? (stg_valid_base + h * 128) : stg_dump_base);
#pragma unroll
                for (int v = 0; v < 8; ++v) ob[v * 8] = d2v[v];
            }
            __builtin_amdgcn_wave_barrier();
        }

        // ---- Coalesced writeout: wave writes 1 KB (2 x 512B bursts of b128),
        // non-temporal (output is streamed, never re-read; keep L2 for the cores).
        float* og = out + (size_t)t * 256;
        v4f o0 = *(v4f*)(ostg + lane * 4);
        v4f o1 = *(v4f*)(ostg + 128 + lane * 4);
        __builtin_nontemporal_store(o0, (v4f*)(og + lane * 4));
        __builtin_nontemporal_store(o1, (v4f*)(og + 128 + lane * 4));
        __builtin_amdgcn_wave_barrier();  // order LDS reads before next token's stores
    }
}

extern "C" void kernel_launch(void* const* d_in, const int* in_sizes, int n_in,
                              void* d_out, int out_size, void* d_ws, size_t ws_size,
                              hipStream_t stream) {
    (void)n_in; (void)out_size; (void)d_ws; (void)ws_size;
    const int*   ids = (const int*)d_in[0];
    const float* cr1 = (const float*)d_in[1];
    const float* cr2 = (const float*)d_in[2];
    const float* cr3 = (const float*)d_in[3];
    const float* ci1 = (const float*)d_in[4];
    const float* ci2 = (const float*)d_in[5];
    const float* ci3 = (const float*)d_in[6];
    float* out = (float*)d_out;
    const int ntok = in_sizes[0];   // B*S = 131072

    // 1024 blocks x 8 waves = 8192 waves; ~16 tokens per wave.
    tt_embed_wmma_kernel<<<1024, 256, 0, stream>>>(ids, cr1, cr2, cr3, ci1, ci2, ci3, out, ntok);
}